// GNNModel_28845000360148
// MI455X (gfx1250) — compile-verified
//
#include <hip/hip_runtime.h>

typedef float v2f __attribute__((ext_vector_type(2)));
typedef float v8f __attribute__((ext_vector_type(8)));

// ---------------- degree / normalization ----------------

__global__ void __launch_bounds__(256) k_deg_init(float* __restrict__ deg, int n) {
  int i = blockIdx.x * 256 + threadIdx.x;
  if (i < n) deg[i] = 1.0f;  // self-loop contributes 1
}

__global__ void __launch_bounds__(256) k_deg_count(const int* __restrict__ dst,
                                                   float* __restrict__ deg, int E) {
  int e = blockIdx.x * 256 + threadIdx.x;
  if (e < E) atomicAdd(&deg[dst[e]], 1.0f);
}

__global__ void __launch_bounds__(256) k_dinv(float* __restrict__ deg, int n) {
  int i = blockIdx.x * 256 + threadIdx.x;
  if (i < n) deg[i] = rsqrtf(fmaxf(deg[i], 1e-12f));
}

// ---------------- GEMM: C[N x 64] = A[N x 64] * W[64 x 64], f32 WMMA ----------------
// One wave computes one 16x16 output tile; 16 chained V_WMMA_F32_16X16X4_F32 over K=64.

__global__ void __launch_bounds__(256) k_gemm64_wmma(const float* __restrict__ A,
                                                     const float* __restrict__ W,
                                                     float* __restrict__ C,
                                                     int nRows, int nTilesM) {
  const int lane  = threadIdx.x & 31;
  const int wave  = threadIdx.x >> 5;
  const int tile  = blockIdx.x * 8 + wave;   // 8 waves per block
  const int tileN = tile & 3;                // 64 cols -> 4 col-tiles
  const int tileM = tile >> 2;
  if (tileM >= nTilesM) return;              // wave-uniform exit (EXEC stays all-1s)

  const int mrow  = lane & 15;
  const int khalf = lane >> 4;               // 0: K pair {0,1}; 1: K pair {2,3}
  int arow = tileM * 16 + mrow;
  if (arow >= nRows) arow = nRows - 1;       // clamp (N is a multiple of 16 here)
  const float* __restrict__ ap = A + (size_t)arow * 64;
  const int ncol = tileN * 16 + (lane & 15);

  v8f acc = {0.f, 0.f, 0.f, 0.f, 0.f, 0.f, 0.f, 0.f};
#pragma unroll
  for (int k0 = 0; k0 < 64; k0 += 4) {
    v2f a, b;
    // A 16x4 fragment: lane-half selects K pair {0,1} vs {2,3}
    a[0] = ap[k0 + 2 * khalf + 0];
    a[1] = ap[k0 + 2 * khalf + 1];
    // B 4x16 fragment: VGPR r holds rows K=r (lanes 0-15) / K=r+2 (lanes 16-31)
    b[0] = W[(size_t)(k0 + 2 * khalf + 0) * 64 + ncol];
    b[1] = W[(size_t)(k0 + 2 * khalf + 1) * 64 + ncol];
    acc = __builtin_amdgcn_wmma_f32_16x16x4_f32(false, a, false, b, (short)0, acc,
                                                false, false);
  }

  // D 16x16: VGPR r -> row r (lanes 0-15) / row r+8 (lanes 16-31)
  const int ccol = tileN * 16 + (lane & 15);
#pragma unroll
  for (int r = 0; r < 8; ++r) {
    int crow = tileM * 16 + r + 8 * khalf;
    if (crow < nRows) C[(size_t)crow * 64 + ccol] = acc[r];
  }
}

// ---------------- aggregation ----------------

// agg[i,:] = tmp[i,:] * dinv[i]^2   (self-loop term; also initializes agg)
__global__ void __launch_bounds__(256) k_selfloop(const float* __restrict__ tmp,
                                                  const float* __restrict__ dinv,
                                                  float* __restrict__ agg, long n64) {
  long idx = (long)blockIdx.x * 256 + threadIdx.x;
  if (idx < n64) {
    long i = idx >> 6;
    float di = dinv[i];
    agg[idx] = tmp[idx] * di * di;
  }
}

// agg[dst,:] += tmp[src,:] * dinv[src]*dinv[dst]  -- one thread per (edge, 4 feats)
__global__ void __launch_bounds__(256) k_scatter(const float* __restrict__ tmp,
                                                 const int* __restrict__ src,
                                                 const int* __restrict__ dst,
                                                 const float* __restrict__ dinv,
                                                 float* __restrict__ agg, long E) {
  long t = (long)blockIdx.x * 256 + threadIdx.x;
  long nwork = E * 16;
  if (t >= nwork) return;
  long e  = t >> 4;
  int  fg = (int)(t & 15) * 4;
  int  s  = src[e];
  int  d  = dst[e];
  float w = dinv[s] * dinv[d];
  const float4 h = *(const float4*)(tmp + (size_t)s * 64 + fg);
  float* op = agg + (size_t)d * 64 + fg;
  atomicAdd(op + 0, h.x * w);
  atomicAdd(op + 1, h.y * w);
  atomicAdd(op + 2, h.z * w);
  atomicAdd(op + 3, h.w * w);
}

// h = relu(agg + b), in place
__global__ void __launch_bounds__(256) k_bias_relu(float* __restrict__ h,
                                                   const float* __restrict__ b, long n64) {
  long idx = (long)blockIdx.x * 256 + threadIdx.x;
  if (idx < n64) h[idx] = fmaxf(h[idx] + b[idx & 63], 0.0f);
}

// ---------------- pooling + FC ----------------

__global__ void k_zero64(float* __restrict__ cs) {
  if (threadIdx.x < 64) cs[threadIdx.x] = 0.0f;
}

// column sums of h[N x 64] into cs[64]; stride (grid*256) divisible by 64 so each
// thread always visits column (tid & 63)
__global__ void __launch_bounds__(256) k_colsum(const float* __restrict__ h,
                                                float* __restrict__ cs, long n64) {
  __shared__ float sm[256];
  float local = 0.0f;
  for (long i = (long)blockIdx.x * 256 + threadIdx.x; i < n64;
       i += (long)gridDim.x * 256)
    local += h[i];
  sm[threadIdx.x] = local;
  __syncthreads();
  if (threadIdx.x < 64)
    atomicAdd(&cs[threadIdx.x], sm[threadIdx.x] + sm[threadIdx.x + 64] +
                                sm[threadIdx.x + 128] + sm[threadIdx.x + 192]);
}

__global__ void k_final(const float* __restrict__ cs, const float* __restrict__ fcw,
                        const float* __restrict__ fcb, float* __restrict__ out,
                        float invN) {
  __shared__ float sm[64];
  int t = threadIdx.x;
  sm[t] = cs[t] * invN * fcw[t];
  __syncthreads();
  if (t < 32) sm[t] += sm[t + 32];
  __syncthreads();
  if (t == 0) {
    float s = 0.0f;
#pragma unroll
    for (int i = 0; i < 32; ++i) s += sm[i];
    out[0] = s + fcb[0];
  }
}

// ---------------- host launch ----------------

extern "C" void kernel_launch(void* const* d_in, const int* in_sizes, int n_in,
                              void* d_out, int out_size, void* d_ws, size_t ws_size,
                              hipStream_t stream) {
  const float* x   = (const float*)d_in[0];
  const int*   ei  = (const int*)d_in[1];   // [2, E] (int32: JAX x64 disabled)
  const float* W1  = (const float*)d_in[3];
  const float* b1  = (const float*)d_in[4];
  const float* W2  = (const float*)d_in[5];
  const float* b2  = (const float*)d_in[6];
  const float* fcw = (const float*)d_in[8];
  const float* fcb = (const float*)d_in[9];

  const int N = in_sizes[0] / 64;
  const int E = in_sizes[1] / 2;
  const int* src = ei;
  const int* dst = ei + E;

  // workspace layout (f32): tmp[N*64] | agg[N*64] | dinv[N] | colsum[64]
  float* tmp  = (float*)d_ws;
  float* agg  = tmp + (size_t)N * 64;
  float* dinv = agg + (size_t)N * 64;
  float* cs   = dinv + N;

  const long n64      = (long)N * 64;
  const int  nTilesM  = (N + 15) / 16;
  const int  gemmBlk  = (nTilesM * 4 + 7) / 8;
  const int  nBlkN    = (N + 255) / 256;
  const int  nBlkE    = (E + 255) / 256;
  const int  nBlkN64  = (int)((n64 + 255) / 256);
  const int  nBlkScat = (int)(((long)E * 16 + 255) / 256);

  // degrees + symmetric normalization (shared by both conv layers)
  k_deg_init<<<nBlkN, 256, 0, stream>>>(dinv, N);
  k_deg_count<<<nBlkE, 256, 0, stream>>>(dst, dinv, E);
  k_dinv<<<nBlkN, 256, 0, stream>>>(dinv, N);

  // layer 1: h1 = relu(Agg(x @ W1) + b1)
  k_gemm64_wmma<<<gemmBlk, 256, 0, stream>>>(x, W1, tmp, N, nTilesM);
  k_selfloop<<<nBlkN64, 256, 0, stream>>>(tmp, dinv, agg, n64);
  k_scatter<<<nBlkScat, 256, 0, stream>>>(tmp, src, dst, dinv, agg, E);
  k_bias_relu<<<nBlkN64, 256, 0, stream>>>(agg, b1, n64);

  // layer 2: h2 = relu(Agg(h1 @ W2) + b2)   (agg holds h1; tmp is free again)
  k_gemm64_wmma<<<gemmBlk, 256, 0, stream>>>(agg, W2, tmp, N, nTilesM);
  k_selfloop<<<nBlkN64, 256, 0, stream>>>(tmp, dinv, agg, n64);
  k_scatter<<<nBlkScat, 256, 0, stream>>>(tmp, src, dst, dinv, agg, E);
  k_bias_relu<<<nBlkN64, 256, 0, stream>>>(agg, b2, n64);

  // global mean pool + fc
  k_zero64<<<1, 64, 0, stream>>>(cs);
  k_colsum<<<512, 256, 0, stream>>>(agg, cs, n64);
  k_final<<<1, 64, 0, stream>>>(cs, fcw, fcb, (float*)d_out, 1.0f / (float)N);
}